// StructuredMaskedCE_42030549958674
// MI455X (gfx1250) — compile-verified
//
#include <hip/hip_runtime.h>
#include <hip/hip_bf16.h>

typedef __attribute__((ext_vector_type(2))) float v2f;
typedef __attribute__((ext_vector_type(8))) float v8f;

#define TILE 16

// ---------------------------------------------------------------------------
// Zero the per-protein accumulators (sum, count) in workspace.
// ---------------------------------------------------------------------------
__global__ void drmsd_zero_ws(float* __restrict__ ws, int n) {
    int i = blockIdx.x * blockDim.x + threadIdx.x;
    if (i < n) ws[i] = 0.0f;
}

// ---------------------------------------------------------------------------
// Main kernel: one wave32 per (protein b, tile_i, tile_j) job, ti <= tj.
// A 16x16 tile of BOTH squared-distance matrices is produced entirely by
// chained V_WMMA_F32_16X16X4_F32 ops:
//   d2 = A1 x B1 + A2 x B2   with A1 row i = ( x, y, z, |x|^2 )
//                                 B1 col j = (-2x,-2y,-2z, 1  )
//                                 A2 row i = ( 1, 0, 0, 0 )
//                                 B2 col j = ( |x|^2, 0, 0, 0 )
// The mask weight matrix w = m_i * m_j is a fifth WMMA (rank-1 outer product).
// Element loop uses native v_sqrt_f32; no cross-lane traffic until the final
// wave reduction.
// ---------------------------------------------------------------------------
__global__ __launch_bounds__(256) void drmsd_tiles(
    const float* __restrict__ pred,     // [B,S,3]
    const float* __restrict__ targ,     // [B,S,3]
    const int*   __restrict__ mask,     // [B,S]
    float*       __restrict__ acc,      // [B,2] = (sum_sq_err, pair_count)
    int B, int S, int T, int P, int totalJobs)
{
    int tid  = blockIdx.x * blockDim.x + threadIdx.x;
    int wave = tid >> 5;           // global wave id == job id (uniform in wave)
    int lane = tid & 31;
    if (wave >= totalJobs) return; // wave-uniform exit; EXEC all-1 below

    int b = wave / P;
    int t = wave - b * P;

    // Map t -> (ti, tj) over upper triangle (incl. diagonal) of T x T tiles.
    float af = (float)(2 * T + 1);
    int ti = (int)((af - __builtin_amdgcn_sqrtf(af * af - 8.0f * (float)t)) * 0.5f);
    if (ti < 0) ti = 0;
    if (ti > T - 1) ti = T - 1;
    while (ti + 1 <= T - 1 && ((ti + 1) * T - ((ti + 1) * ti) / 2) <= t) ti++;
    while (ti > 0 && (ti * T - (ti * (ti - 1)) / 2) > t) ti--;
    int off = ti * T - (ti * (ti - 1)) / 2;
    int tj = ti + (t - off);

    int p    = lane & 15;   // point-within-tile (row M side for A, col N for B)
    int half = lane >> 4;   // 0: K=0,1 half ; 1: K=2,3 half

    int giRaw = ti * TILE + p;
    int gjRaw = tj * TILE + p;
    int gi = giRaw < S ? giRaw : S - 1;   // clamped for loads only
    int gj = gjRaw < S ? gjRaw : S - 1;

    const float* pb = pred + (size_t)b * S * 3;
    const float* tb = targ + (size_t)b * S * 3;
    const int*   mb = mask + (size_t)b * S;

    float xpi = pb[gi * 3 + 0], ypi = pb[gi * 3 + 1], zpi = pb[gi * 3 + 2];
    float xpj = pb[gj * 3 + 0], ypj = pb[gj * 3 + 1], zpj = pb[gj * 3 + 2];
    float xti = tb[gi * 3 + 0], yti = tb[gi * 3 + 1], zti = tb[gi * 3 + 2];
    float xtj = tb[gj * 3 + 0], ytj = tb[gj * 3 + 1], ztj = tb[gj * 3 + 2];

    // Squared norms, per lane (lanes L and L+16 carry the same point p).
    float sqpi = xpi * xpi + ypi * ypi + zpi * zpi;
    float sqpj = xpj * xpj + ypj * ypj + zpj * zpj;
    float sqti = xti * xti + yti * yti + zti * zti;
    float sqtj = xtj * xtj + ytj * ytj + ztj * ztj;

    // Masks, branchless (clamped load + select keeps EXEC uniform).
    float mi = ((giRaw < S) && (mb[gi] != 0)) ? 1.0f : 0.0f;
    float mj = ((gjRaw < S) && (mb[gj] != 0)) ? 1.0f : 0.0f;

    float hf = (float)half;             // 0 for lanes 0-15, 1 for lanes 16-31
    float lo = 1.0f - hf;               // 1 on lanes 0-15

    // A1/B1: coordinate + norm packing (K = 0,1 on lanes 0-15; K = 2,3 above).
    v2f A1p, B1p, A1t, B1t;
    A1p.x = half ? zpi : xpi;            A1p.y = half ? sqpi : ypi;
    B1p.x = -2.0f * (half ? zpj : xpj);  B1p.y = half ? 1.0f : (-2.0f * ypj);
    A1t.x = half ? zti : xti;            A1t.y = half ? sqti : yti;
    B1t.x = -2.0f * (half ? ztj : xtj);  B1t.y = half ? 1.0f : (-2.0f * ytj);

    // A2 = ones column (K=0 only); B2 = |xj|^2 row (K=0 only).
    v2f A2;  A2.x = lo;          A2.y = 0.0f;
    v2f B2p; B2p.x = lo * sqpj;  B2p.y = 0.0f;
    v2f B2t; B2t.x = lo * sqtj;  B2t.y = 0.0f;

    // Mask outer product operands (K=0 only).
    v2f Am;  Am.x = lo * mi;     Am.y = 0.0f;
    v2f Bm;  Bm.x = lo * mj;     Bm.y = 0.0f;

    v8f czero = {};
    // d2 tiles via chained WMMA accumulation (D of first feeds C of second).
    v8f d2p = __builtin_amdgcn_wmma_f32_16x16x4_f32(
        false, A1p, false, B1p, (short)0, czero, false, false);
    d2p = __builtin_amdgcn_wmma_f32_16x16x4_f32(
        false, A2, false, B2p, (short)0, d2p, false, false);

    v8f d2t = __builtin_amdgcn_wmma_f32_16x16x4_f32(
        false, A1t, false, B1t, (short)0, czero, false, false);
    d2t = __builtin_amdgcn_wmma_f32_16x16x4_f32(
        false, A2, false, B2t, (short)0, d2t, false, false);

    // Mask weight tile w[M][N] = m_i[M] * m_j[N].
    v8f wm = __builtin_amdgcn_wmma_f32_16x16x4_f32(
        false, Am, false, Bm, (short)0, czero, false, false);

    bool tileOffDiag = (ti < tj);       // wave-uniform: every i < j in tile

    float sum = 0.0f, cnt = 0.0f;
#pragma unroll
    for (int v = 0; v < 8; ++v) {
        int M = half * 8 + v;           // row of accumulator element v
        // Strict upper triangle: uniform true off-diagonal, M < N on diagonal.
        float tri = (tileOffDiag || (M < p)) ? 1.0f : 0.0f;
        float w = wm[v] * tri;
        float dp = __builtin_amdgcn_sqrtf(fmaxf(d2p[v], 0.0f));
        float dt = __builtin_amdgcn_sqrtf(fmaxf(d2t[v], 0.0f));
        float diff = dp - dt;
        sum += w * diff * diff;
        cnt += w;
    }

    // Wave32 reduction.
#pragma unroll
    for (int o = 16; o > 0; o >>= 1) {
        sum += __shfl_xor(sum, o, 32);
        cnt += __shfl_xor(cnt, o, 32);
    }
    if (lane == 0) {
        atomicAdd(&acc[2 * b + 0], sum);
        atomicAdd(&acc[2 * b + 1], cnt);
    }
}

// ---------------------------------------------------------------------------
// Finalize: mean over proteins of sqrt(seg_mean + 1e-6).
// ---------------------------------------------------------------------------
__global__ void drmsd_finalize(const float* __restrict__ acc,
                               float* __restrict__ out, int B) {
    if (blockIdx.x == 0 && threadIdx.x == 0) {
        float s = 0.0f;
        for (int b = 0; b < B; ++b) {
            float m = acc[2 * b + 0] / acc[2 * b + 1];
            s += sqrtf(m + 1e-6f);
        }
        out[0] = s / (float)B;
    }
}

extern "C" void kernel_launch(void* const* d_in, const int* in_sizes, int n_in,
                              void* d_out, int out_size, void* d_ws, size_t ws_size,
                              hipStream_t stream) {
    const float* pred = (const float*)d_in[0];   // [B,S,3] f32
    const float* targ = (const float*)d_in[1];   // [B,S,3] f32
    const int*   mask = (const int*)d_in[2];     // [B,S]   i32
    float* out = (float*)d_out;
    float* acc = (float*)d_ws;                   // [B,2] accumulators

    const int B = 8;                             // per reference setup
    const int S = in_sizes[2] / B;               // 2400
    const int T = (S + TILE - 1) / TILE;         // 150 tiles
    const int P = T * (T + 1) / 2;               // tile pairs per protein
    const int totalJobs = B * P;

    drmsd_zero_ws<<<1, 64, 0, stream>>>(acc, 2 * B);

    const int threads = 256;                     // 8 wave32 per block
    const int wavesPerBlock = threads / 32;
    const int blocks = (totalJobs + wavesPerBlock - 1) / wavesPerBlock;
    drmsd_tiles<<<blocks, threads, 0, stream>>>(pred, targ, mask, acc,
                                                B, S, T, P, totalJobs);

    drmsd_finalize<<<1, 32, 0, stream>>>(acc, out, B);
}